// GcnPolicy_77403900609164
// MI455X (gfx1250) — compile-verified
//
#include <hip/hip_runtime.h>
#include <hip/hip_bf16.h>

typedef __attribute__((ext_vector_type(2))) float v2f;
typedef __attribute__((ext_vector_type(8))) float v8f;

#define TPB 256

// ---------------- normalization precompute ----------------

__global__ void k_fill(float* __restrict__ p, float v, int n) {
  int i = blockIdx.x * blockDim.x + threadIdx.x;
  if (i < n) p[i] = v;
}

__global__ void k_deg(const int* __restrict__ dst, float* __restrict__ deg, int E) {
  int e = blockIdx.x * blockDim.x + threadIdx.x;
  if (e < E) atomicAdd(&deg[dst[e]], 1.0f);
}

__global__ void k_rsqrt_inplace(float* __restrict__ d, int n) {
  int i = blockIdx.x * blockDim.x + threadIdx.x;
  if (i < n) d[i] = rsqrtf(d[i]);   // deg >= 1 guaranteed by self-loop init
}

__global__ void k_norm(const int* __restrict__ src, const int* __restrict__ dst,
                       const float* __restrict__ dis, float* __restrict__ norm, int E) {
  int e = blockIdx.x * blockDim.x + threadIdx.x;
  if (e < E) norm[e] = dis[src[e]] * dis[dst[e]];
}

// ---------------- operand packing for WMMA ----------------

// Pack x [N,2] -> xpad [N,4] with zero cols 2,3 so A fragments load unguarded.
__global__ void k_pack_x(const float* __restrict__ x, float* __restrict__ xp, int n4) {
  int i = blockIdx.x * blockDim.x + threadIdx.x;
  if (i < n4) {
    int k = i & 3;
    xp[i] = (k < 2) ? x[(i >> 2) * 2 + k] : 0.f;
  }
}

// Pack W [FI,8] into the B-fragment pair layout, zero-padded to 8x16:
//   Bp[(k>>1)*32 + col*2 + (k&1)] = (k<FI && col<8) ? W[k*8+col] : 0
// so lane l's fragment for even k0 is one contiguous 8-byte load.
__global__ void k_packW(const float* __restrict__ W, float* __restrict__ Bp, int FI) {
  int i = threadIdx.x;                 // one block of 128 threads
  if (i < 128) {
    int sub = i & 1, col = (i >> 1) & 15, p = i >> 5;
    int k = 2 * p + sub;
    Bp[i] = (k < FI && col < 8) ? W[k * 8 + col] : 0.f;
  }
}

// ---------------- dense h @ W via V_WMMA_F32_16X16X4_F32 ----------------
// One wave32 computes a full 16-node x 16-col tile (cols 8..15 padded zero).
// A (16x4 f32): VGPR0 = K {0,2} across lane halves, VGPR1 = K {1,3} -> one v2f
// load per lane per chunk.  B identical K split, N = lane&15, from packed Bp.
// D is staged through LDS (wave-in-order DS) and stored as one b128 per lane.
template <int KP>   // padded K / input row stride: 4 (layer0) or 8
__global__ void k_gemm_wmma(const float* __restrict__ in, const float* __restrict__ Bp,
                            float* __restrict__ t, int n_full) {
  __shared__ __align__(16) float sm[TPB / 32][128];
  int wv   = threadIdx.x >> 5;
  int wave = (blockIdx.x * blockDim.x + threadIdx.x) >> 5;
  int lane = threadIdx.x & 31;
  if (wave >= n_full) return;          // wave-uniform: EXEC all-1s at the WMMA
  int hi = lane >> 4;
  int l  = lane & 15;
  int base = wave * 16;
  int row  = base + l;
  v8f c = {};
#pragma unroll
  for (int kb = 0; kb < KP; kb += 4) {
    int k0 = kb + 2 * hi;              // even; k1 = k0+1 is the adjacent float
    v2f a = *(const v2f*)(in + (size_t)row * KP + k0);
    v2f b = *(const v2f*)(Bp + (k0 >> 1) * 32 + l * 2);
    c = __builtin_amdgcn_wmma_f32_16x16x4_f32(false, a, false, b, (short)0, c,
                                              false, false);
  }
  // D: VGPR j -> row j (lanes 0-15) / row j+8 (lanes 16-31), col = lane&15.
  if (l < 8) {
#pragma unroll
    for (int j = 0; j < 8; ++j) sm[wv][(j + 8 * hi) * 8 + l] = c[j];
  }
  float4 o = *(const float4*)(&sm[wv][lane * 4]);
  *(float4*)(t + (size_t)base * 8 + lane * 4) = o;
}

// Scalar tail for n % 16 != 0 (not hit for N=100000, kept for generality).
__global__ void k_gemm_tail(const float* __restrict__ in, const float* __restrict__ W,
                            float* __restrict__ t, int start, int n, int stride, int FI) {
  int idx = blockIdx.x * blockDim.x + threadIdx.x;
  int total = (n - start) * 8;
  if (idx >= total) return;
  int row = start + (idx >> 3), col = idx & 7;
  float s = 0.f;
  for (int k = 0; k < FI; ++k) s += in[(size_t)row * stride + k] * W[k * 8 + col];
  t[(size_t)row * 8 + col] = s;
}

// ---------------- edge gather/scatter (L2-atomic bound) ----------------

__global__ void k_selfloop(const float* __restrict__ t, const float* __restrict__ dis,
                           float* __restrict__ acc, int n8) {
  int i = blockIdx.x * blockDim.x + threadIdx.x;
  if (i < n8) {
    float d = dis[i >> 3];
    acc[i] = t[i] * d * d;             // also initializes the accumulator
  }
}

__global__ void k_scatter(const int* __restrict__ src, const int* __restrict__ dst,
                          const float* __restrict__ norm, const float* __restrict__ t,
                          float* __restrict__ acc, int E) {
  int e = blockIdx.x * blockDim.x + threadIdx.x;
  if (e >= E) return;
  // CDNA5 global_prefetch_b8 ahead of the streaming edge reads (speculative-safe)
  __builtin_prefetch(src + e + 16384, 0, 1);
  __builtin_prefetch(dst + e + 16384, 0, 1);
  __builtin_prefetch(norm + e + 16384, 0, 1);
  int s = src[e], d = dst[e];
  float w = norm[e];
  const float4* tp4 = reinterpret_cast<const float4*>(t + (size_t)s * 8);
  float4 a = tp4[0], b = tp4[1];
  float* o = acc + (size_t)d * 8;
  atomicAdd(o + 0, a.x * w); atomicAdd(o + 1, a.y * w);
  atomicAdd(o + 2, a.z * w); atomicAdd(o + 3, a.w * w);
  atomicAdd(o + 4, b.x * w); atomicAdd(o + 5, b.y * w);
  atomicAdd(o + 6, b.z * w); atomicAdd(o + 7, b.w * w);
}

__global__ void k_bias_elu(const float* __restrict__ acc, const float* __restrict__ bias,
                           float* __restrict__ h, int n8) {
  int i = blockIdx.x * blockDim.x + threadIdx.x;
  if (i < n8) {
    float v = acc[i] + bias[i & 7];
    h[i] = v > 0.f ? v : (expf(v) - 1.f);   // ELU alpha=1
  }
}

// ---------------- heads ----------------

__global__ void k_heads_dot(const float* __restrict__ h, const float* __restrict__ Wv,
                            const float* __restrict__ Wp, float* __restrict__ tv,
                            float* __restrict__ tp, int n) {
  int i = blockIdx.x * blockDim.x + threadIdx.x;
  if (i >= n) return;
  const float* hi = h + (size_t)i * 8;
  float sv = 0.f, sp = 0.f;
#pragma unroll
  for (int f = 0; f < 8; ++f) { sv += hi[f] * Wv[f]; sp += hi[f] * Wp[f]; }
  tv[i] = sv; tp[i] = sp;
}

__global__ void k_heads_self(const float* __restrict__ tv, const float* __restrict__ tp,
                             const float* __restrict__ dis, const float* __restrict__ bv,
                             float* __restrict__ vacc, float* __restrict__ pacc,
                             float* __restrict__ out, int n) {
  int i = blockIdx.x * blockDim.x + threadIdx.x;
  if (i == 0) out[n] = bv[0];          // init value slot each call (deterministic)
  if (i < n) {
    float d = dis[i]; float d2 = d * d;
    vacc[i] = tv[i] * d2;
    pacc[i] = tp[i] * d2;
  }
}

__global__ void k_heads_scatter(const int* __restrict__ src, const int* __restrict__ dst,
                                const float* __restrict__ norm, const float* __restrict__ tv,
                                const float* __restrict__ tp, float* __restrict__ vacc,
                                float* __restrict__ pacc, int E) {
  int e = blockIdx.x * blockDim.x + threadIdx.x;
  if (e >= E) return;
  int s = src[e], d = dst[e];
  float w = norm[e];
  atomicAdd(&vacc[d], tv[s] * w);
  atomicAdd(&pacc[d], tp[s] * w);
}

__global__ void k_finalize(const float* __restrict__ vacc, const float* __restrict__ pacc,
                           const float* __restrict__ bp, float* __restrict__ out, int n) {
  __shared__ float sm[TPB];
  int i = blockIdx.x * blockDim.x + threadIdx.x;
  float v = 0.f;
  if (i < n) { out[i] = pacc[i] + bp[0]; v = vacc[i]; }
  sm[threadIdx.x] = v;
  __syncthreads();
  for (int s = TPB / 2; s > 0; s >>= 1) {
    if ((int)threadIdx.x < s) sm[threadIdx.x] += sm[threadIdx.x + s];
    __syncthreads();
  }
  if (threadIdx.x == 0) atomicAdd(out + n, sm[0] * (1.0f / (float)n));
}

// ---------------- driver ----------------

static inline size_t pad4(size_t x) { return (x + 3) & ~(size_t)3; }  // 16B align

extern "C" void kernel_launch(void* const* d_in, const int* in_sizes, int n_in,
                              void* d_out, int out_size, void* d_ws, size_t ws_size,
                              hipStream_t stream) {
  (void)n_in; (void)out_size; (void)ws_size;
  const float* x  = (const float*)d_in[0];
  const int*   ei = (const int*)d_in[1];
  const float* W0 = (const float*)d_in[2];
  const float* b0 = (const float*)d_in[3];
  const float* W1 = (const float*)d_in[4];
  const float* b1 = (const float*)d_in[5];
  const float* W2 = (const float*)d_in[6];
  const float* b2 = (const float*)d_in[7];
  const float* W3 = (const float*)d_in[8];
  const float* b3 = (const float*)d_in[9];
  const float* Wv = (const float*)d_in[10];
  const float* bv = (const float*)d_in[11];
  const float* Wp = (const float*)d_in[12];
  const float* bp = (const float*)d_in[13];
  float* out = (float*)d_out;

  const int N  = in_sizes[0] / 2;   // x is [N, 2]
  const int E  = in_sizes[1] / 2;   // edge_index is [2, E]
  const int N8 = N * 8;

  const int* src = ei;
  const int* dst = ei + E;

  float* ws   = (float*)d_ws;
  float* dis  = ws;  ws += pad4(N);
  float* norm = ws;  ws += pad4(E);
  float* t    = ws;  ws += pad4(N8);
  float* acc  = ws;  ws += pad4(N8);
  float* h    = ws;  ws += pad4(N8);
  float* xpad = ws;  ws += pad4((size_t)N * 4);
  float* Bp   = ws;  ws += 128;
  float* tv   = ws;  ws += pad4(N);
  float* tp   = ws;  ws += pad4(N);
  float* vacc = ws;  ws += pad4(N);
  float* pacc = ws;  ws += pad4(N);

  dim3 blk(TPB);
  int gN  = (N  + TPB - 1) / TPB;
  int gN8 = (N8 + TPB - 1) / TPB;
  int gN4 = (N * 4 + TPB - 1) / TPB;
  int gE  = (E  + TPB - 1) / TPB;
  int n_full = N / 16;
  int tail_start = n_full * 16;
  int tail = N - tail_start;
  int gT = (n_full + (TPB / 32) - 1) / (TPB / 32);

  // degree (with self-loop) -> dis = deg^-1/2 -> per-edge norm
  k_fill<<<gN, blk, 0, stream>>>(dis, 1.0f, N);
  k_deg<<<gE, blk, 0, stream>>>(dst, dis, E);
  k_rsqrt_inplace<<<gN, blk, 0, stream>>>(dis, N);
  k_norm<<<gE, blk, 0, stream>>>(src, dst, dis, norm, E);

  // layer-0 input padded to stride 4 for unguarded A-fragment loads
  k_pack_x<<<gN4, blk, 0, stream>>>(x, xpad, N * 4);

  const float* Ws[4] = {W0, W1, W2, W3};
  const float* bs[4] = {b0, b1, b2, b3};
  const float* cur = xpad;
  for (int layer = 0; layer < 4; ++layer) {
    const int FI = (layer == 0) ? 2 : 8;
    const int stride = (layer == 0) ? 4 : 8;
    k_packW<<<1, 128, 0, stream>>>(Ws[layer], Bp, FI);
    if (layer == 0)
      k_gemm_wmma<4><<<gT, blk, 0, stream>>>(cur, Bp, t, n_full);
    else
      k_gemm_wmma<8><<<gT, blk, 0, stream>>>(cur, Bp, t, n_full);
    if (tail > 0) {
      int gTail = (tail * 8 + TPB - 1) / TPB;
      k_gemm_tail<<<gTail, blk, 0, stream>>>(cur, Ws[layer], t, tail_start, N, stride, FI);
    }
    k_selfloop<<<gN8, blk, 0, stream>>>(t, dis, acc, N8);
    k_scatter<<<gE, blk, 0, stream>>>(src, dst, norm, t, acc, E);
    k_bias_elu<<<gN8, blk, 0, stream>>>(acc, bs[layer], h, N8);
    cur = h;
  }

  k_heads_dot<<<gN, blk, 0, stream>>>(h, Wv, Wp, tv, tp, N);
  k_heads_self<<<gN, blk, 0, stream>>>(tv, tp, dis, bv, vacc, pacc, out, N);
  k_heads_scatter<<<gE, blk, 0, stream>>>(src, dst, norm, tv, tp, vacc, pacc, E);
  k_finalize<<<gN, blk, 0, stream>>>(vacc, pacc, bp, out, N);
}